// m1_40810779246941
// MI455X (gfx1250) — compile-verified
//
#include <hip/hip_runtime.h>
#include <stdint.h>

// h_t = a*h_{t-1} + b*x_t + c over T steps; z = w*h_T + e
//   h_T = b * S + c * G,  S = sum_t a^(T-1-t) x_t,  G = (1 - a^T)/(1 - a)
#define T_TOTAL   33554432            // 2^25
#define NBLK      2048                // partial blocks (d_ws holds NBLK floats = 8 KB)
#define BT        256                 // threads per block (8 waves of 32)
#define SEG       (T_TOTAL / NBLK)    // 16384 floats per block
#define STG       (BT * 8)            // 2048 floats per stage (2 x b128 per thread)
#define NSTG      (SEG / STG)         // 8 stages per block
#define HALF      1024                // offset of second quad within a stage (floats)

typedef __attribute__((ext_vector_type(2))) float v2f;
typedef __attribute__((ext_vector_type(4))) float v4f;
typedef __attribute__((ext_vector_type(8))) float v8f;
typedef int v4i __attribute__((vector_size(16)));   // matches builtin's V4i

#if __has_builtin(__builtin_amdgcn_global_load_async_to_lds_b128)
#define HAVE_ASYNC 1
#else
#define HAVE_ASYNC 0
#endif

#if HAVE_ASYNC
typedef __attribute__((address_space(1))) v4i gv4i;  // global int4
typedef __attribute__((address_space(3))) v4i lv4i;  // LDS int4

__device__ __forceinline__ void async_copy16(const float* g, float* l) {
  // global -> LDS 16B per lane via the CDNA5 async data path (tracked by ASYNCcnt)
  __builtin_amdgcn_global_load_async_to_lds_b128(
      (gv4i*)(uintptr_t)g,
      (lv4i*)(uint32_t)(uintptr_t)l,
      0, 0);
}
template <int N> __device__ __forceinline__ void wait_async() {
#if __has_builtin(__builtin_amdgcn_s_wait_asynccnt)
  __builtin_amdgcn_s_wait_asynccnt(N);
#else
  asm volatile("s_wait_asynccnt %0" ::"i"(N) : "memory");
#endif
}
#endif

// Kernel 1: per-block geometrically-weighted partial sums.
// Block bid computes  a^(SEG*(NBLK-1-bid)) * sum_{o<SEG} a^(SEG-1-o) * x[bid*SEG+o]
__global__ __launch_bounds__(BT) void scan_partial(const float* __restrict__ x,
                                                   const float* __restrict__ pa,
                                                   float* __restrict__ partial) {
  __shared__ float buf[2 * STG];   // double-buffered async staging (16 KB)
  __shared__ float red[BT];

  const int bid = blockIdx.x;
  const int t = threadIdx.x;
  const float a = pa[0];
  const float la = logf(a);

  // Backward stage iteration: start weights at the LAST stage of this segment.
  //   quad A element o = 4t+k        -> weight a^((STG-1) - (4t+k))
  //   quad B element o = HALF+4t+k   -> weight a^((STG-1) - HALF - (4t+k))
  const float a1024 = expf(la * (float)HALF);                   // a^1024
  float u3 = expf(la * (float)(HALF - 1 - (4 * t + 3)));        // a^(1020-4t)
  float u2 = u3 * a;
  float u1 = u2 * a;
  float u0 = u1 * a;
  float w3 = u3 * a1024, w2 = u2 * a1024, w1 = u1 * a1024, w0 = u0 * a1024;
  const float m = a1024 * a1024;                                // a^STG per backward stage

  const float* gx = x + (size_t)bid * SEG;
  float acc = 0.0f;

#if HAVE_ASYNC
  {  // prologue: prefetch last stage into half 0
    const float* g = gx + (size_t)(NSTG - 1) * STG;
    async_copy16(g + 4 * t, &buf[4 * t]);
    async_copy16(g + HALF + 4 * t, &buf[HALF + 4 * t]);
  }
  int cur = 0;
#pragma unroll
  for (int n = NSTG - 1; n >= 0; --n) {
    if (n > 0) {  // prefetch next (older) stage into the other half
      const float* g = gx + (size_t)(n - 1) * STG;
      float* l = &buf[(cur ^ 1) * STG];
      async_copy16(g + 4 * t, l + 4 * t);
      async_copy16(g + HALF + 4 * t, l + HALF + 4 * t);
      wait_async<2>();  // oldest two (current stage) complete
    } else {
      wait_async<0>();
    }
    asm volatile("" ::: "memory");
    const float* lb = &buf[cur * STG];
    v4f va = *(const v4f*)(lb + 4 * t);
    v4f vb = *(const v4f*)(lb + HALF + 4 * t);
    acc = fmaf(va.x, w0, acc); acc = fmaf(va.y, w1, acc);
    acc = fmaf(va.z, w2, acc); acc = fmaf(va.w, w3, acc);
    acc = fmaf(vb.x, u0, acc); acc = fmaf(vb.y, u1, acc);
    acc = fmaf(vb.z, u2, acc); acc = fmaf(vb.w, u3, acc);
    w0 *= m; w1 *= m; w2 *= m; w3 *= m;
    u0 *= m; u1 *= m; u2 *= m; u3 *= m;
    asm volatile("" ::: "memory");
    cur ^= 1;
  }
#else
#pragma unroll
  for (int n = NSTG - 1; n >= 0; --n) {
    const float* g = gx + (size_t)n * STG;
    v4f va = __builtin_nontemporal_load((const v4f*)(g + 4 * t));
    v4f vb = __builtin_nontemporal_load((const v4f*)(g + HALF + 4 * t));
    acc = fmaf(va.x, w0, acc); acc = fmaf(va.y, w1, acc);
    acc = fmaf(va.z, w2, acc); acc = fmaf(va.w, w3, acc);
    acc = fmaf(vb.x, u0, acc); acc = fmaf(vb.y, u1, acc);
    acc = fmaf(vb.z, u2, acc); acc = fmaf(vb.w, u3, acc);
    w0 *= m; w1 *= m; w2 *= m; w3 *= m;
    u0 *= m; u1 *= m; u2 *= m; u3 *= m;
  }
#endif

  // deterministic fixed-order block reduction
  red[t] = acc;
  __syncthreads();
#pragma unroll
  for (int s = BT / 2; s > 0; s >>= 1) {
    if (t < s) red[t] += red[t + s];
    __syncthreads();
  }
  if (t == 0) {
    const float scale = expf(la * (float)(SEG * (NBLK - 1 - bid)));  // underflow->0 ok
    partial[bid] = red[0] * scale;
  }
}

// Kernel 2: deterministic reduction of NBLK partials via fp32 WMMA (exact f32 adds:
// A = all-ones 16x4, so D[m][n] = sum_k B[k][n]; every partial lands in exactly one
// (k,n) slot, so summing row M=0 over N gives the total), then the closed form.
__global__ __launch_bounds__(32) void finalize(const float* __restrict__ partial,
                                               const float* __restrict__ pa,
                                               const float* __restrict__ pb,
                                               const float* __restrict__ pc,
                                               const float* __restrict__ pw,
                                               const float* __restrict__ pe,
                                               float* __restrict__ out) {
  __shared__ float tmp[32];
  const int lane = threadIdx.x;
  float lanesum;
#if __has_builtin(__builtin_amdgcn_wmma_f32_16x16x4_f32)
  v8f acc = {};
  const v2f ones = {1.0f, 1.0f};
#pragma unroll
  for (int i = 0; i < NBLK / 64; ++i) {  // 64 partials per WMMA, 32 iterations
    v2f bm;
    bm.x = partial[i * 64 + lane];
    bm.y = partial[i * 64 + 32 + lane];
    acc = __builtin_amdgcn_wmma_f32_16x16x4_f32(false, ones, false, bm,
                                                (short)0, acc, false, false);
  }
  // C/D layout: VGPR0, lanes 0..15 hold M=0, N=lane -> 16 column sums
  lanesum = (lane < 16) ? acc[0] : 0.0f;
#else
  float s = 0.0f;
  for (int i = lane; i < NBLK; i += 32) s += partial[i];
  lanesum = s;
#endif
  tmp[lane] = lanesum;
  __syncthreads();
  if (lane == 0) {
    float S = 0.0f;
    for (int n = 0; n < 32; ++n) S += tmp[n];
    const float a = pa[0], b = pb[0], c = pc[0], w = pw[0], e = pe[0];
    const float la = logf(a);
    const float aT = expf(la * (float)T_TOTAL);  // underflows to 0 for these a
    const float G = (1.0f - aT) / (1.0f - a);    // sum_{k<T} a^k
    out[0] = w * (b * S + c * G) + e;
  }
}

extern "C" void kernel_launch(void* const* d_in, const int* in_sizes, int n_in,
                              void* d_out, int out_size, void* d_ws, size_t ws_size,
                              hipStream_t stream) {
  (void)in_sizes; (void)n_in; (void)out_size; (void)ws_size;
  const float* x  = (const float*)d_in[0];
  const float* pa = (const float*)d_in[1];
  const float* pb = (const float*)d_in[2];
  const float* pc = (const float*)d_in[3];
  const float* pw = (const float*)d_in[4];
  const float* pe = (const float*)d_in[5];
  float* partial = (float*)d_ws;  // NBLK floats = 8 KB scratch

  scan_partial<<<NBLK, BT, 0, stream>>>(x, pa, partial);
  finalize<<<1, 32, 0, stream>>>(partial, pa, pb, pc, pw, pe, (float*)d_out);
}